// CompactBilinearPooling_10728828305722
// MI455X (gfx1250) — compile-verified
//
#include <hip/hip_runtime.h>
#include <math.h>

// CompactBilinearPooling fused kernel for gfx1250 (MI455X).
// One workgroup per batch row:
//   - count-sketch x,y into ZR/ZI (LDS ds_add_f32 atomics): z = xs + i*ys
//   - one forward length-8192 FFT (DIF radix 16,16,16,2) in LDS using
//     v_wmma_f32_16x16x4_f32 (12 WMMAs per 16x16 complex matmul, Karatsuba)
//   - Hermitian split + spectral product in digit-reversed order
//   - inverse FFT (exact stage inverses), write real part once to HBM
// Twiddles come from a 32KB half-wave LDS table (no per-element sincos).
// HBM traffic ~192MB total -> ~8us at 23.3 TB/s; compute kept on WMMA path.

typedef __attribute__((ext_vector_type(2))) float v2f;
typedef __attribute__((ext_vector_type(8))) float v8f;

#define NFFT   8192
#define BLOCK  256
#define NWAVES (BLOCK / 32)
#define TWO_PI 6.283185307179586f

// ---- f32 WMMA wrapper: D(16x16) += A(16x4) * B(4x16) -----------------------
__device__ __forceinline__ v8f wmma_f32(v2f a, v2f b, v8f c) {
  // 8 args: (neg_a, A, neg_b, B, c_mod, C, reuse_a, reuse_b)
  return __builtin_amdgcn_wmma_f32_16x16x4_f32(false, a, false, b, (short)0, c,
                                               false, false);
}

// ---- W_8192^t from half-wave LDS table (W^{t+4096} = -W^t) -----------------
template <bool CONJ>
__device__ __forceinline__ void tw_get(const float* __restrict__ TW, int t,
                                       float& c, float& s) {
  const int   idx = t & (NFFT / 2 - 1);
  const float sg  = (t & (NFFT / 2)) ? -1.0f : 1.0f;
  const float cc  = TW[2 * idx];
  const float ss  = TW[2 * idx + 1];
  c = cc * sg;
  s = CONJ ? (-ss * sg) : (ss * sg);
}

// ---- digit-reversed position of natural frequency k (radices 16,16,16,2) --
__device__ __forceinline__ int pos_of_k(int k) {
  const int r1 = k & 15;
  const int r2 = (k >> 4) & 15;
  const int r3 = (k >> 8) & 15;
  const int r4 = (k >> 12) & 1;
  return (r1 << 9) | (r2 << 5) | (r3 << 1) | r4;
}

// ---- radix-2 stage (L == 2, trivial twiddles; self-inverse up to scale) ----
__device__ __forceinline__ void radix2_stage(float* __restrict__ re,
                                             float* __restrict__ im) {
  for (int p = threadIdx.x; p < NFFT / 2; p += BLOCK) {
    const int i0 = 2 * p, i1 = i0 + 1;
    const float a_r = re[i0], a_i = im[i0];
    const float b_r = re[i1], b_i = im[i1];
    re[i0] = a_r + b_r; im[i0] = a_i + b_i;
    re[i1] = a_r - b_r; im[i1] = a_i - b_i;
  }
}

// ---- radix-16 stage via WMMA ----------------------------------------------
// Forward (INV=false): v = DFT16(u); out[m] = v[m] * W_L^{j*m}   (DIF)
// Inverse (INV=true):  u[m] *= conj(W_L^{j*m}); out = conj(DFT16)(u)
// 512 butterflies/stage; 16 butterflies = one 16x16 complex matmul done as
// 3 real matmuls (Karatsuba) = 12 chained v_wmma_f32_16x16x4_f32.
template <bool INV>
__device__ __forceinline__ void radix16_stage(float* __restrict__ re,
                                              float* __restrict__ im,
                                              const float* __restrict__ TW,
                                              const int logL,
                                              const v2f* __restrict__ ar,
                                              const v2f* __restrict__ ai) {
  const int logH = logL - 4;
  const int h    = 1 << logH;          // butterfly stride
  const int sh   = 13 - logL;          // twiddle index scale: W_L = W_N^{N/L}
  const int lane = threadIdx.x & 31;
  const int wave = threadIdx.x >> 5;
  const int hi   = lane >> 4;          // half-wave select (0/1)
  const int col  = lane & 15;          // this lane's B/D column

  // Signed imaginary part of the DFT16 operand (conjugated for inverse) and
  // the Karatsuba sum operand; both constant across the whole stage.
  v2f aiS[4], aS[4];
#pragma unroll
  for (int c = 0; c < 4; ++c) {
    if (INV) aiS[c] = -ai[c]; else aiS[c] = ai[c];
    aS[c] = ar[c] + aiS[c];
  }

  for (int set = wave; set < 32; set += NWAVES) {   // 32 matmul batches/stage
    const int t    = set * 16 + col;     // flat butterfly index (per lane)
    const int blk  = t >> logH;
    const int j    = t & (h - 1);
    const int base = (blk << logL) + j;  // element m lives at base + m*h

    // ---- load B fragments (rows interleaved like the ISA A-layout) ----
    v2f br[4], bi[4], bs[4];
#pragma unroll
    for (int c = 0; c < 4; ++c) {
#pragma unroll
      for (int v = 0; v < 2; ++v) {
        const int m   = 4 * c + v + 2 * hi;       // K row of this slot
        const int idx = base + (m << logH);
        float xr = re[idx];
        float xi = im[idx];
        if (INV) {  // pre-twiddle with conj(W_L^{j*m})
          float cw, sw;
          tw_get<true>(TW, (j * m) << sh, cw, sw);
          const float tr = xr * cw - xi * sw;
          const float ti = xr * sw + xi * cw;
          xr = tr; xi = ti;
        }
        br[c][v] = xr;
        bi[c][v] = xi;
      }
      bs[c] = br[c] + bi[c];
    }

    // ---- complex 16x16x16 matmul: 3 real matmuls x 4 K-chunks ----
    v8f m1 = {};
    v8f m2 = {};
    v8f m3 = {};
#pragma unroll
    for (int c = 0; c < 4; ++c) m1 = wmma_f32(ar[c],  br[c], m1);
#pragma unroll
    for (int c = 0; c < 4; ++c) m2 = wmma_f32(aiS[c], bi[c], m2);
#pragma unroll
    for (int c = 0; c < 4; ++c) m3 = wmma_f32(aS[c],  bs[c], m3);
    const v8f cr = m1 - m2;
    const v8f ci = m3 - m1 - m2;

    // ---- twiddle (forward) + writeback; D: M = v + 8*hi, N = col ----
#pragma unroll
    for (int v = 0; v < 8; ++v) {
      const int m   = v + 8 * hi;
      const int idx = base + (m << logH);
      float xr = cr[v];
      float xi = ci[v];
      if (!INV) {  // post-twiddle W_L^{j*m}
        float cw, sw;
        tw_get<false>(TW, (j * m) << sh, cw, sw);
        const float tr = xr * cw - xi * sw;
        const float ti = xr * sw + xi * cw;
        xr = tr; xi = ti;
      }
      re[idx] = xr;
      im[idx] = xi;
    }
  }
}

__global__ __launch_bounds__(BLOCK) void cbp_fused_kernel(
    const float* __restrict__ x, const float* __restrict__ y,
    const int* __restrict__ h1, const int* __restrict__ h2,
    const float* __restrict__ s1, const float* __restrict__ s2,
    float* __restrict__ out, int D1, int D2) {
  extern __shared__ float lds[];           // ZR | ZI | TW  = 96 KB
  float* ZR = lds;
  float* ZI = lds + NFFT;
  float* TW = lds + 2 * NFFT;              // 4096 (re,im) pairs

  const int row = blockIdx.x;
  const int tid = threadIdx.x;
  const float* xrow = x + (size_t)row * D1;
  const float* yrow = y + (size_t)row * D2;
  __builtin_prefetch(xrow, 0, 0);          // global_prefetch_b8
  __builtin_prefetch(yrow, 0, 0);

  // ---- zero sketch bins, build half-wave twiddle table ----
  for (int i = tid; i < NFFT; i += BLOCK) {
    ZR[i] = 0.0f;
    ZI[i] = 0.0f;
  }
  for (int t = tid; t < NFFT / 2; t += BLOCK) {
    const float ang = -TWO_PI * (float)t * (1.0f / (float)NFFT);
    float sn, cs; __sincosf(ang, &sn, &cs);
    TW[2 * t]     = cs;
    TW[2 * t + 1] = sn;
  }
  __syncthreads();

  // ---- count-sketch scatter: z = xs + i*ys (LDS ds_add_f32 atomics) ----
  for (int i = tid; i < D1; i += BLOCK) atomicAdd(&ZR[h1[i]], xrow[i] * s1[i]);
  for (int i = tid; i < D2; i += BLOCK) atomicAdd(&ZI[h2[i]], yrow[i] * s2[i]);
  __syncthreads();

  // ---- DFT16 A-matrix fragments (W16^{m*k}) per ISA VGPR layout ----
  v2f ar[4], ai[4];
  {
    const int lane = tid & 31;
    const int hi   = lane >> 4;
    const int mA   = lane & 15;            // A's M index (output row)
#pragma unroll
    for (int c = 0; c < 4; ++c) {
#pragma unroll
      for (int v = 0; v < 2; ++v) {
        const int k = 4 * c + v + 2 * hi;  // A's K index (ISA layout)
        const float ang = -TWO_PI * (float)((mA * k) & 15) * (1.0f / 16.0f);
        float sn, cs; __sincosf(ang, &sn, &cs);
        ar[c][v] = cs;
        ai[c][v] = sn;
      }
    }
  }

  // ---- forward FFT of z (DIF: L = 8192, 512, 32, then radix-2) ----
  radix16_stage<false>(ZR, ZI, TW, 13, ar, ai);
  __syncthreads();
  radix16_stage<false>(ZR, ZI, TW, 9, ar, ai);
  __syncthreads();
  radix16_stage<false>(ZR, ZI, TW, 5, ar, ai);
  __syncthreads();
  radix2_stage(ZR, ZI);
  __syncthreads();

  // ---- Hermitian split + spectral product, in digit-reversed order ----
  // Xf = (Z[k]+conj Z[N-k])/2 ; Yf = (Z[k]-conj Z[N-k])/(2i) ; P = Xf*Yf.
  // Both spectra Hermitian => P[N-k] = conj(P[k]); each k<=N/2 owns the
  // disjoint position pair (p, pp), so in-place update is race-free.
  for (int k = tid; k <= NFFT / 2; k += BLOCK) {
    const int kp = (NFFT - k) & (NFFT - 1);
    const int p  = pos_of_k(k);
    const int pp = pos_of_k(kp);
    const float x1 = ZR[p],  y1 = ZI[p];
    const float x2 = ZR[pp], y2 = ZI[pp];
    const float ar_ = 0.5f * (x1 + x2);
    const float ai_ = 0.5f * (y1 - y2);
    const float br_ = 0.5f * (y1 + y2);
    const float bi_ = -0.5f * (x1 - x2);
    const float pr = ar_ * br_ - ai_ * bi_;
    const float pi = ar_ * bi_ + ai_ * br_;
    ZR[p] = pr;   ZI[p] = pi;
    ZR[pp] = pr;  ZI[pp] = -pi;   // self-paired k (0, N/2) has pi == 0
  }
  __syncthreads();

  // ---- inverse FFT (exact stage inverses, reverse order) ----
  radix2_stage(ZR, ZI);
  __syncthreads();
  radix16_stage<true>(ZR, ZI, TW, 5, ar, ai);
  __syncthreads();
  radix16_stage<true>(ZR, ZI, TW, 9, ar, ai);
  __syncthreads();
  radix16_stage<true>(ZR, ZI, TW, 13, ar, ai);
  __syncthreads();

  // ---- real part; fold all 1/16^3 * 1/2 inverse scales into 1/8192 ----
  float* orow = out + (size_t)row * NFFT;
  for (int i = tid; i < NFFT; i += BLOCK) {
    orow[i] = ZR[i] * (1.0f / (float)NFFT);
  }
}

extern "C" void kernel_launch(void* const* d_in, const int* in_sizes, int n_in,
                              void* d_out, int out_size, void* d_ws,
                              size_t ws_size, hipStream_t stream) {
  (void)n_in; (void)d_ws; (void)ws_size; (void)out_size;
  const float* x  = (const float*)d_in[0];
  const float* y  = (const float*)d_in[1];
  const int*   h1 = (const int*)d_in[2];
  const int*   h2 = (const int*)d_in[3];
  const float* s1 = (const float*)d_in[4];
  const float* s2 = (const float*)d_in[5];
  float*       out = (float*)d_out;

  const int D1 = in_sizes[2];
  const int D2 = in_sizes[3];
  const int B  = (D1 > 0) ? in_sizes[0] / D1 : 0;

  const size_t shmem = (size_t)3 * NFFT * sizeof(float);  // 96 KB
  hipLaunchKernelGGL(cbp_fused_kernel, dim3(B), dim3(BLOCK), shmem, stream,
                     x, y, h1, h2, s1, s2, out, D1, D2);
}